// InfectionGNGuidedBP_28647431864466
// MI455X (gfx1250) — compile-verified
//
#include <hip/hip_runtime.h>

#define N_NODES   1000000
#define N_EDGES   16000000
#define N_GRAPHS  64

// workspace layout (in floats)
#define N1_OFF    0          // n1  : N_NODES*8 = 8,000,000 floats (32 MB, L2-resident)
#define AGG_OFF   8000000    // agg : N_NODES*8 = 8,000,000 floats (32 MB, L2-resident)
#define G_OFF     16000000   // g   : N_GRAPHS*8 = 512 floats

typedef float v2f __attribute__((ext_vector_type(2)));
typedef float v4f __attribute__((ext_vector_type(4)));
typedef float v8f __attribute__((ext_vector_type(8)));

// D(16x16,f32) = A(16x4,f32) x B(4x16,f32) + C   -- V_WMMA_F32_16X16X4_F32
__device__ __forceinline__ v8f wmma_f32_16x16x4(v2f a, v2f b, v8f c) {
  return __builtin_amdgcn_wmma_f32_16x16x4_f32(false, a, false, b, (short)0, c,
                                               false, false);
}

__global__ __launch_bounds__(256) void zero_kernel(float* __restrict__ p, int count) {
  int i = blockIdx.x * blockDim.x + threadIdx.x;
  if (i < count) p[i] = 0.0f;
}

// ---------------------------------------------------------------------------
// Kernel A: n1 = relu(node_feats @ Wn1.T + bn1), computed transposed via WMMA:
//   D = Wn1(16x4, rows 8..15 zero) x node_feats.T(4x16)  -> lane n (lanes 0..15)
//   holds node n's 8 output features in acc[0..7].
// ---------------------------------------------------------------------------
__global__ __launch_bounds__(256) void node_enc_kernel(
    const float* __restrict__ node_feats, const float* __restrict__ Wn1,
    const float* __restrict__ bn1, float* __restrict__ n1) {
  const int lane = threadIdx.x & 31;
  const int row  = lane & 15;        // A: M-row / B: N-column index
  const int hi   = lane >> 4;        // selects K-pair {0,1} vs {2,3}
  const int wid    = (blockIdx.x * blockDim.x + threadIdx.x) >> 5;
  const int nwaves = (gridDim.x * blockDim.x) >> 5;

  // A-matrix (loop invariant): a.x = A[row][2*hi], a.y = A[row][2*hi+1]
  v2f a;
  a.x = (row < 8) ? Wn1[row * 4 + 2 * hi]     : 0.0f;
  a.y = (row < 8) ? Wn1[row * 4 + 2 * hi + 1] : 0.0f;

  float bias[8];
#pragma unroll
  for (int f = 0; f < 8; ++f) bias[f] = bn1[f];

  const int ntiles = N_NODES / 16;   // 62,500 exact
  for (int t = wid; t < ntiles; t += nwaves) {
    const int node = t * 16 + row;
    // B column `row`: b.x = feats[node][2*hi], b.y = feats[node][2*hi+1]
    const float* p = node_feats + (size_t)node * 4 + 2 * hi;
    v2f b; b.x = p[0]; b.y = p[1];

    v8f acc = {0.f, 0.f, 0.f, 0.f, 0.f, 0.f, 0.f, 0.f};
    acc = wmma_f32_16x16x4(a, b, acc);

    if (hi == 0) {  // lanes 0..15 hold M=0..7 (the 8 real output features)
      float o[8];
#pragma unroll
      for (int f = 0; f < 8; ++f) o[f] = fmaxf(acc[f] + bias[f], 0.0f);
      v4f* dst = (v4f*)(n1 + (size_t)node * 8);
      v4f lo = {o[0], o[1], o[2], o[3]};
      v4f hv = {o[4], o[5], o[6], o[7]};
      dst[0] = lo;
      dst[1] = hv;
    }
  }
}

// ---------------------------------------------------------------------------
// Kernel B: per-edge 2-layer MLP + scatter-add into agg[receiver].
// Memory-bound: stream edge arrays non-temporally (keep n1/agg hot in L2),
// gather n1[sender] (L2 hit), 8 non-returning f32 atomics per edge.
// ---------------------------------------------------------------------------
__global__ __launch_bounds__(256) void edge_kernel(
    const float* __restrict__ edge_feats, const int* __restrict__ senders,
    const int* __restrict__ receivers,
    const float* __restrict__ We1, const float* __restrict__ be1,
    const float* __restrict__ We2, const float* __restrict__ Ws2,
    const float* __restrict__ be2,
    const float* __restrict__ n1, float* __restrict__ agg) {
  float we1[8], b1[4], we2[32], ws2[64], b2[8];
#pragma unroll
  for (int i = 0; i < 8;  ++i) we1[i] = We1[i];
#pragma unroll
  for (int i = 0; i < 4;  ++i) b1[i]  = be1[i];
#pragma unroll
  for (int i = 0; i < 32; ++i) we2[i] = We2[i];
#pragma unroll
  for (int i = 0; i < 64; ++i) ws2[i] = Ws2[i];
#pragma unroll
  for (int i = 0; i < 8;  ++i) b2[i]  = be2[i];

  const int stride = gridDim.x * blockDim.x;
  for (int e = blockIdx.x * blockDim.x + threadIdx.x; e < N_EDGES; e += stride) {
    v2f ef = __builtin_nontemporal_load(&((const v2f*)edge_feats)[e]);
    int s  = __builtin_nontemporal_load(senders + e);
    int r  = __builtin_nontemporal_load(receivers + e);

    const v4f* sp = (const v4f*)(n1 + (size_t)s * 8);
    v4f s0 = sp[0];
    v4f s1 = sp[1];

    float e1v[4];
#pragma unroll
    for (int j = 0; j < 4; ++j)
      e1v[j] = fmaxf(we1[2 * j] * ef.x + we1[2 * j + 1] * ef.y + b1[j], 0.0f);

#pragma unroll
    for (int j = 0; j < 8; ++j) {
      float t = b2[j];
      t += e1v[0] * we2[4 * j + 0] + e1v[1] * we2[4 * j + 1] +
           e1v[2] * we2[4 * j + 2] + e1v[3] * we2[4 * j + 3];
      t += s0.x * ws2[8 * j + 0] + s0.y * ws2[8 * j + 1] +
           s0.z * ws2[8 * j + 2] + s0.w * ws2[8 * j + 3] +
           s1.x * ws2[8 * j + 4] + s1.y * ws2[8 * j + 5] +
           s1.z * ws2[8 * j + 6] + s1.w * ws2[8 * j + 7];
      t = fmaxf(t, 0.0f);
      atomicAdd(&agg[(size_t)r * 8 + j], t);  // non-returning -> L2 RMW
    }
  }
}

// ---------------------------------------------------------------------------
// Kernel C: n2 = relu(Wn2 @ n1.T + Win2 @ agg.T + bn2) via 4 chained
// 16x16x4 WMMAs (K=8 split into two K=4 chunks per weight matrix), then
// node readout (dot with Wro) and per-block LDS graph accumulation.
// ---------------------------------------------------------------------------
__global__ __launch_bounds__(256) void node_upd_kernel(
    const float* __restrict__ n1, const float* __restrict__ agg,
    const int* __restrict__ node_graph,
    const float* __restrict__ Wn2, const float* __restrict__ Win2,
    const float* __restrict__ bn2,
    const float* __restrict__ Wro, const float* __restrict__ bro,
    float* __restrict__ out_nodes, float* __restrict__ g_part) {
  __shared__ float lds_g[N_GRAPHS * 8];
  for (int i = threadIdx.x; i < N_GRAPHS * 8; i += blockDim.x) lds_g[i] = 0.0f;
  __syncthreads();

  const int lane = threadIdx.x & 31;
  const int row  = lane & 15;
  const int hi   = lane >> 4;
  const int wid    = (blockIdx.x * blockDim.x + threadIdx.x) >> 5;
  const int nwaves = (gridDim.x * blockDim.x) >> 5;

  // Loop-invariant A chunks: chunk c covers K = 4c..4c+3
  v2f aN[2], aI[2];
#pragma unroll
  for (int c = 0; c < 2; ++c) {
    int k = 4 * c + 2 * hi;
    aN[c].x = (row < 8) ? Wn2[row * 8 + k]      : 0.0f;
    aN[c].y = (row < 8) ? Wn2[row * 8 + k + 1]  : 0.0f;
    aI[c].x = (row < 8) ? Win2[row * 8 + k]     : 0.0f;
    aI[c].y = (row < 8) ? Win2[row * 8 + k + 1] : 0.0f;
  }
  float bias[8], wro[8];
#pragma unroll
  for (int f = 0; f < 8; ++f) { bias[f] = bn2[f]; wro[f] = Wro[f]; }
  const float brob = bro[0];

  const int ntiles = N_NODES / 16;
  for (int t = wid; t < ntiles; t += nwaves) {
    const int node = t * 16 + row;
    const float* p1 = n1  + (size_t)node * 8;
    const float* pa = agg + (size_t)node * 8;

    v8f acc = {0.f, 0.f, 0.f, 0.f, 0.f, 0.f, 0.f, 0.f};
#pragma unroll
    for (int c = 0; c < 2; ++c) {
      int k = 4 * c + 2 * hi;
      v2f b; b.x = p1[k]; b.y = p1[k + 1];
      acc = wmma_f32_16x16x4(aN[c], b, acc);
    }
#pragma unroll
    for (int c = 0; c < 2; ++c) {
      int k = 4 * c + 2 * hi;
      v2f b; b.x = pa[k]; b.y = pa[k + 1];
      acc = wmma_f32_16x16x4(aI[c], b, acc);
    }

    if (hi == 0) {
      int   gi = node_graph[node];
      float ro = brob;
#pragma unroll
      for (int f = 0; f < 8; ++f) {
        float v = fmaxf(acc[f] + bias[f], 0.0f);
        ro += v * wro[f];
        atomicAdd(&lds_g[gi * 8 + f], v);  // ds_add_f32
      }
      out_nodes[node] = ro;
    }
  }

  __syncthreads();
  for (int i = threadIdx.x; i < N_GRAPHS * 8; i += blockDim.x)
    atomicAdd(&g_part[i], lds_g[i]);
}

// ---------------------------------------------------------------------------
// Kernel D: globals_out = g @ Wg.T + bg   (64 rows, trivial)
// ---------------------------------------------------------------------------
__global__ void globals_kernel(const float* __restrict__ g,
                               const float* __restrict__ Wg,
                               const float* __restrict__ bg,
                               float* __restrict__ out) {
  int i = threadIdx.x;
  if (i < N_GRAPHS) {
    const float* gr = g + i * 8;
    float t = bg[0];
#pragma unroll
    for (int f = 0; f < 8; ++f) t += gr[f] * Wg[f];
    out[N_NODES + i] = t;
  }
}

extern "C" void kernel_launch(void* const* d_in, const int* in_sizes, int n_in,
                              void* d_out, int out_size, void* d_ws, size_t ws_size,
                              hipStream_t stream) {
  const float* node_feats = (const float*)d_in[0];
  const float* edge_feats = (const float*)d_in[1];
  const int*   senders    = (const int*)d_in[2];
  const int*   receivers  = (const int*)d_in[3];
  const int*   node_graph = (const int*)d_in[4];
  const float* We1  = (const float*)d_in[5];
  const float* be1  = (const float*)d_in[6];
  const float* Wn1  = (const float*)d_in[7];
  const float* bn1  = (const float*)d_in[8];
  const float* We2  = (const float*)d_in[9];
  const float* Ws2  = (const float*)d_in[10];
  const float* be2  = (const float*)d_in[11];
  const float* Wn2  = (const float*)d_in[12];
  const float* Win2 = (const float*)d_in[13];
  const float* bn2  = (const float*)d_in[14];
  const float* Wro  = (const float*)d_in[15];
  const float* bro  = (const float*)d_in[16];
  const float* Wg   = (const float*)d_in[17];
  const float* bg   = (const float*)d_in[18];

  float* ws  = (float*)d_ws;
  float* n1  = ws + N1_OFF;
  float* agg = ws + AGG_OFF;
  float* gws = ws + G_OFF;
  float* out = (float*)d_out;

  // 1) zero agg + g (contiguous in ws)
  {
    int count  = N_NODES * 8 + N_GRAPHS * 8;
    int blocks = (count + 255) / 256;
    zero_kernel<<<blocks, 256, 0, stream>>>(agg, count);
  }
  // 2) node encoder (WMMA f32 16x16x4)
  node_enc_kernel<<<512, 256, 0, stream>>>(node_feats, Wn1, bn1, n1);
  // 3) edge MLP + scatter (memory-bound: NT streaming + L2 atomics)
  edge_kernel<<<16384, 256, 0, stream>>>(edge_feats, senders, receivers,
                                         We1, be1, We2, Ws2, be2, n1, agg);
  // 4) node update (4x chained WMMA) + node readout + graph partial sums
  node_upd_kernel<<<512, 256, 0, stream>>>(n1, agg, node_graph, Wn2, Win2, bn2,
                                           Wro, bro, out, gws);
  // 5) globals readout
  globals_kernel<<<1, 64, 0, stream>>>(gws, Wg, bg, out);
}